// GATLayer_3736621547953
// MI455X (gfx1250) — compile-verified
//
#include <hip/hip_runtime.h>
#include <math.h>

typedef float v2f __attribute__((ext_vector_type(2)));
typedef float v8f __attribute__((ext_vector_type(8)));
typedef int   v4i __attribute__((ext_vector_type(4)));

#define N_NODES 8192
#define F_IN    256
#define F_OUT   64
#define NEG_BIG (-9000000000000000.0f)
#define LRELU_A 0.2f
#define JT      32                   // columns per flash tile
#define LDSTR   36                   // padded LDS stride: float4-aligned + conflict-free
#define SPLITS  4                    // column-range split for occupancy
#define JCOLS   (N_NODES / SPLITS)   // 2048 columns per split
#define TILES   (JCOLS / JT)         // 64 tiles per split (even)

// ---- CDNA5 async global->LDS path (guarded; fallback keeps same structure) ----
#if defined(__has_builtin)
#  if __has_builtin(__builtin_amdgcn_global_load_async_to_lds_b128)
#    define ASYNC_LDS 1
#  else
#    define ASYNC_LDS 0
#  endif
#else
#  define ASYNC_LDS 0
#endif

#if ASYNC_LDS
typedef __attribute__((address_space(1))) v4i glb_v4i;   // clang prints AS(1) as __device__
typedef __attribute__((address_space(3))) v4i lds_v4i;
#endif

__device__ __forceinline__ void stage_issue(float* lds_dst, const float* gsrc, float4* regs) {
#if ASYNC_LDS
    // 32B per thread via two async B128 DMAs (ASYNCcnt-tracked, no VGPR data)
    __builtin_amdgcn_global_load_async_to_lds_b128((glb_v4i*)gsrc,       (lds_v4i*)lds_dst,       0, 0);
    __builtin_amdgcn_global_load_async_to_lds_b128((glb_v4i*)(gsrc + 4), (lds_v4i*)(lds_dst + 4), 0, 0);
    (void)regs;
#else
    const float4* s = (const float4*)gsrc;
    regs[0] = s[0];
    regs[1] = s[1];
    (void)lds_dst;
#endif
}

__device__ __forceinline__ void stage_commit(float* lds_dst, const float4* regs) {
#if !ASYNC_LDS
    float4* d = (float4*)lds_dst;
    d[0] = regs[0];
    d[1] = regs[1];
#else
    (void)lds_dst; (void)regs;
#endif
}

__device__ __forceinline__ void stage_wait() {
#if ASYNC_LDS
#  if __has_builtin(__builtin_amdgcn_s_wait_asynccnt)
    __builtin_amdgcn_s_wait_asynccnt(0);
#  else
    asm volatile("s_wait_asynccnt 0x0" ::: "memory");
#  endif
#endif
}

// ---------------------------------------------------------------------------
// Kernel 1: hT[f][n] = sum_k input[n][k] * W[k][f]  (stored feature-major)
// One wave = one 16x16 tile, 64 chained v_wmma_f32_16x16x4_f32 (K=256).
// ---------------------------------------------------------------------------
__global__ __launch_bounds__(256) void gat_gemm_hT(const float* __restrict__ inp,
                                                   const float* __restrict__ W,
                                                   float* __restrict__ hT) {
    const int lane = threadIdx.x & 31;
    const int wave = threadIdx.x >> 5;
    const int tile = blockIdx.x * 8 + wave;      // 2048 tiles
    const int rowbase = (tile >> 2) * 16;
    const int fbase   = (tile & 3) * 16;
    const int lr   = lane & 15;
    const int half = lane >> 4;
    const int koff = half * 2;

    const float* arow = inp + (size_t)(rowbase + lr) * F_IN + koff;
    const float* bcol = W + fbase + lr;

    v8f c = {};
    #pragma unroll 8
    for (int s = 0; s < F_IN / 4; ++s) {
        const int k = 4 * s;
        v2f a = *(const v2f*)(arow + k);
        v2f b;
        b.x = bcol[(size_t)(k + koff)     * F_OUT];
        b.y = bcol[(size_t)(k + koff + 1) * F_OUT];
        c = __builtin_amdgcn_wmma_f32_16x16x4_f32(false, a, false, b,
                                                  (short)0, c, false, false);
    }
    float* outcol = hT + (size_t)(fbase + lr) * N_NODES + rowbase + 8 * half;
    #pragma unroll
    for (int r = 0; r < 8; ++r) outcol[r] = c[r];
}

// ---------------------------------------------------------------------------
// Kernel 2: s1/s2 projections; hT is feature-major -> fully coalesced.
// ---------------------------------------------------------------------------
__global__ __launch_bounds__(256) void gat_scores(const float* __restrict__ hT,
                                                  const float* __restrict__ a,
                                                  float* __restrict__ s1,
                                                  float* __restrict__ s2) {
    const int i = blockIdx.x * 256 + threadIdx.x;
    float acc1 = 0.f, acc2 = 0.f;
    #pragma unroll 8
    for (int f = 0; f < F_OUT; ++f) {
        const float hv = hT[(size_t)f * N_NODES + i];
        acc1 = fmaf(hv, a[f],         acc1);
        acc2 = fmaf(hv, a[F_OUT + f], acc2);
    }
    s1[i] = acc1;
    s2[i] = acc2;
}

// ---------------------------------------------------------------------------
// Kernel 3: flash partial over one column split. Double-buffered LDS h tiles
// (async-to-LDS DMA), register-pipelined adj/s2 stream, online softmax,
// P.h via v_wmma_f32_16x16x4_f32. Writes unnormalized (acc, m, l) partials.
// ---------------------------------------------------------------------------
__global__ __launch_bounds__(256) void gat_flash_part(const int*   __restrict__ adj,
                                                      const float* __restrict__ hT,
                                                      const float* __restrict__ s1,
                                                      const float* __restrict__ s2,
                                                      float* __restrict__ pacc,
                                                      float* __restrict__ pm,
                                                      float* __restrict__ pl) {
    __shared__ float lds_h[2][F_OUT * LDSTR];    // 2 x 9216 B

    const int lane = threadIdx.x & 31;
    const int wave = threadIdx.x >> 5;
    const int split    = blockIdx.x & 3;
    const int rowgroup = blockIdx.x >> 2;        // 64 row groups
    const int rowbase  = (rowgroup * 8 + wave) * 16;
    const int jbase    = split * JCOLS;
    const int lr   = lane & 15;
    const int half = lane >> 4;
    const int koff = half * 2;

    float m = -INFINITY, l = 0.f;
    v8f acc0 = {}, acc1 = {}, acc2 = {}, acc3 = {};

    const float  s1r  = s1[rowbase + lr];
    const size_t arow = (size_t)(rowbase + lr) * N_NODES;

    // staging assignment: thread -> (feature tf, 8-col chunk tq)
    const int tf = threadIdx.x >> 2;
    const int tq = (threadIdx.x & 3) * 8;
    const float* hsrc = hT + (size_t)tf * N_NODES + tq;
    float* ldst0 = &lds_h[0][tf * LDSTR + tq];
    float* ldst1 = &lds_h[1][tf * LDSTR + tq];
    const float* lh0 = &lds_h[0][0];
    const float* lh1 = &lds_h[1][0];

    auto load_adj = [&](int2* ad, float2* sv, int c0) {
        #pragma unroll
        for (int s = 0; s < 8; ++s) {
            const int c = c0 + 4 * s + koff;                    // even -> 8B aligned
            ad[s] = *(const int2*)(adj + arow + c);
            sv[s] = *(const float2*)(s2 + c);
        }
    };
    auto pf = [&](int c0) {
        if (c0 < jbase + JCOLS)
            __builtin_prefetch(adj + arow + c0 + lane, 0, 0);   // global_prefetch_b8
    };
    auto compute_tile = [&](const int2* ad, const float2* sv, const float* hb) {
        float p[16];
        float tmax = -INFINITY;
        #pragma unroll
        for (int s = 0; s < 8; ++s) {
            float e0 = s1r + sv[s].x; e0 = e0 > 0.f ? e0 : LRELU_A * e0;
            float e1 = s1r + sv[s].y; e1 = e1 > 0.f ? e1 : LRELU_A * e1;
            e0 = ad[s].x > 0 ? e0 : NEG_BIG;
            e1 = ad[s].y > 0 ? e1 : NEG_BIG;
            p[2 * s] = e0; p[2 * s + 1] = e1;
            tmax = fmaxf(tmax, fmaxf(e0, e1));
        }
        tmax = fmaxf(tmax, __shfl_xor(tmax, 16, 32));           // merge lane halves
        const float newm  = fmaxf(m, tmax);
        const float scale = __expf(m - newm);                   // 0 on first tile
        float psum = 0.f;
        #pragma unroll
        for (int k = 0; k < 16; ++k) { p[k] = __expf(p[k] - newm); psum += p[k]; }
        psum += __shfl_xor(psum, 16, 32);
        l = l * scale + psum;
        m = newm;
        #pragma unroll
        for (int r = 0; r < 8; ++r) {                           // row-scale broadcast
            const float sc = __shfl(scale, r + (half << 3), 32);
            acc0[r] *= sc; acc1[r] *= sc; acc2[r] *= sc; acc3[r] *= sc;
        }
        #pragma unroll
        for (int s = 0; s < 8; ++s) {                           // P(16x32) @ h(32x64)
            v2f A; A.x = p[2 * s]; A.y = p[2 * s + 1];
            const int cc = 4 * s + koff;
            v2f b0 = *(const v2f*)(hb + (lr     ) * LDSTR + cc);
            acc0 = __builtin_amdgcn_wmma_f32_16x16x4_f32(false, A, false, b0,
                                                         (short)0, acc0, false, false);
            v2f b1 = *(const v2f*)(hb + (16 + lr) * LDSTR + cc);
            acc1 = __builtin_amdgcn_wmma_f32_16x16x4_f32(false, A, false, b1,
                                                         (short)0, acc1, false, false);
            v2f b2 = *(const v2f*)(hb + (32 + lr) * LDSTR + cc);
            acc2 = __builtin_amdgcn_wmma_f32_16x16x4_f32(false, A, false, b2,
                                                         (short)0, acc2, false, false);
            v2f b3 = *(const v2f*)(hb + (48 + lr) * LDSTR + cc);
            acc3 = __builtin_amdgcn_wmma_f32_16x16x4_f32(false, A, false, b3,
                                                         (short)0, acc3, false, false);
        }
    };

    float4 sregA[2], sregB[2];
    int2   adA[8],  adB[8];
    float2 svA[8],  svB[8];

    // prologue: tile 0 staged + adj tile 0 in registers
    stage_issue(ldst0, hsrc + jbase, sregA);
    stage_commit(ldst0, sregA);
    load_adj(adA, svA, jbase);

    for (int t = 0; t < TILES; t += 2) {
        const int j0 = jbase + t * JT;

        stage_wait();                                 // tile t DMA complete
        __syncthreads();
        stage_issue(ldst1, hsrc + j0 + JT, sregB);    // tile t+1 -> buf1 (overlaps)
        load_adj(adB, svB, j0 + JT);                  // adj pipeline 1 tile ahead
        pf(j0 + 2 * JT);
        compute_tile(adA, svA, lh0);
        stage_commit(ldst1, sregB);

        stage_wait();                                 // tile t+1 DMA complete
        __syncthreads();
        const bool more = (t + 2 < TILES);
        if (more) {
            stage_issue(ldst0, hsrc + j0 + 2 * JT, sregA);
            load_adj(adA, svA, j0 + 2 * JT);
        }
        pf(j0 + 3 * JT);
        compute_tile(adB, svB, lh1);
        if (more) stage_commit(ldst0, sregA);
    }

    // write unnormalized partials
    #pragma unroll
    for (int r = 0; r < 8; ++r) {
        const size_t row = (size_t)(split * N_NODES + rowbase + r + 8 * half);
        float* po = pacc + row * F_OUT + lr;
        po[0]  = acc0[r];
        po[16] = acc1[r];
        po[32] = acc2[r];
        po[48] = acc3[r];
    }
    if (lane < 16) {
        pm[split * N_NODES + rowbase + lr] = m;
        pl[split * N_NODES + rowbase + lr] = l;
    }
}

// ---------------------------------------------------------------------------
// Kernel 4: exact online-softmax merge of the 4 column-split partials + ELU.
// ---------------------------------------------------------------------------
__global__ __launch_bounds__(256) void gat_merge(const float* __restrict__ pacc,
                                                 const float* __restrict__ pm,
                                                 const float* __restrict__ pl,
                                                 float* __restrict__ out) {
    const int idx = blockIdx.x * 256 + threadIdx.x;   // 0 .. N*F_OUT-1
    const int i = idx >> 6;
    const int f = idx & 63;
    const float m0 = pm[i];
    const float m1 = pm[N_NODES + i];
    const float m2 = pm[2 * N_NODES + i];
    const float m3 = pm[3 * N_NODES + i];
    const float M  = fmaxf(fmaxf(m0, m1), fmaxf(m2, m3));
    const float w0 = __expf(m0 - M), w1 = __expf(m1 - M);
    const float w2 = __expf(m2 - M), w3 = __expf(m3 - M);
    const float L  = pl[i] * w0 + pl[N_NODES + i] * w1
                   + pl[2 * N_NODES + i] * w2 + pl[3 * N_NODES + i] * w3;
    const float num = pacc[(size_t)i * F_OUT + f]                    * w0
                    + pacc[((size_t)N_NODES     + i) * F_OUT + f]    * w1
                    + pacc[((size_t)2 * N_NODES + i) * F_OUT + f]    * w2
                    + pacc[((size_t)3 * N_NODES + i) * F_OUT + f]    * w3;
    const float v = num / L;
    out[idx] = v > 0.f ? v : __expf(v) - 1.f;          // ELU (alpha=1)
}

// ---------------------------------------------------------------------------
extern "C" void kernel_launch(void* const* d_in, const int* in_sizes, int n_in,
                              void* d_out, int out_size, void* d_ws, size_t ws_size,
                              hipStream_t stream) {
    const float* inp = (const float*)d_in[0];   // 8192 x 256
    const int*   adj = (const int*)  d_in[1];   // 8192 x 8192
    const float* W   = (const float*)d_in[2];   // 256 x 64
    const float* a   = (const float*)d_in[3];   // 128 x 1
    float* out = (float*)d_out;                 // 8192 x 64

    float* hT  = (float*)d_ws;                          // 2 MB
    float* s1  = hT  + (size_t)F_OUT * N_NODES;         // 32 KB
    float* s2v = s1  + N_NODES;                         // 32 KB
    float* pm  = s2v + N_NODES;                         // 128 KB
    float* pl  = pm  + (size_t)SPLITS * N_NODES;        // 128 KB
    float* pacc= pl  + (size_t)SPLITS * N_NODES;        // 8 MB

    gat_gemm_hT   <<<256,  256, 0, stream>>>(inp, W, hT);
    gat_scores    <<<32,   256, 0, stream>>>(hT, a, s1, s2v);
    gat_flash_part<<<256,  256, 0, stream>>>(adj, hT, s1, s2v, pacc, pm, pl);
    gat_merge     <<<2048, 256, 0, stream>>>(pacc, pm, pl, out);
}